// MinLSTM_59657095741645
// MI455X (gfx1250) — compile-verified
//
#include <hip/hip_runtime.h>
#include <math.h>

// Problem constants (from reference): B=8, S=4096, D=512, H=2*D=1024
#define BB 8
#define SS 4096
#define DD 512
#define HH 1024
#define MM (BB * SS)        // 32768 rows (flattened b,s)
#define NC 64               // scan chunks per sequence
#define CSZ 64              // steps per chunk (NC*CSZ == SS)

typedef __attribute__((ext_vector_type(2))) float v2f;
typedef __attribute__((ext_vector_type(8))) float v8f;

__device__ __forceinline__ float softplusf(float x) {
    return fmaxf(x, 0.0f) + log1pf(expf(-fabsf(x)));
}
__device__ __forceinline__ float logaddexpf_(float a, float b) {
    float m = fmaxf(a, b);
    if (m == -__builtin_inff()) return -__builtin_inff();
    return m + log1pf(expf(-fabsf(a - b)));
}
__device__ __forceinline__ float log_g(float x) {
    return (x >= 0.0f) ? logf(x + 0.5f) : -softplusf(-x);
}

// ---------------------------------------------------------------------------
// Kernel 1: proj = inputs @ W_proj^T with fused gate epilogue.
// Per wave: MT1=2 row tiles (32 rows) x NT1=2 col tiles x 3 gates
//   -> 12 accumulators, 12 WMMAs per k-step fed by 2 A + 6 B loads.
// One-stage software pipeline: prefetch k+4 fragments during current WMMAs.
// 32-bit offsets from uniform bases -> saddr-form global loads.
// ---------------------------------------------------------------------------
#define MT1 2
#define NT1 2
__global__ __launch_bounds__(128) void minlstm_gemm1_gates(
    const float* __restrict__ Ain,   // (MM, DD)
    const float* __restrict__ Wp,    // (3*HH, DD)
    float* __restrict__ wsF,
    float* __restrict__ wsC)
{
    const int lane  = threadIdx.x & 31;
    const int wave  = blockIdx.x * (blockDim.x >> 5) + (threadIdx.x >> 5);
    const int rowGrp = wave & (MM / (16 * MT1) - 1);   // 1024 row groups
    const int ng     = wave >> 10;                     // 0..31 col groups
    const int row0  = rowGrp * (16 * MT1);
    const int lrow  = lane & 15;
    const int lhalf = lane >> 4;

    unsigned aoff[MT1];
#pragma unroll
    for (int m = 0; m < MT1; ++m)
        aoff[m] = (unsigned)(row0 + m * 16 + lrow) * DD + 2 * lhalf;

    int nbase[NT1];
    unsigned boff[3 * NT1];
#pragma unroll
    for (int j = 0; j < NT1; ++j) {
        nbase[j] = ng * (16 * NT1) + j * 16;
#pragma unroll
        for (int g = 0; g < 3; ++g)
            boff[g * NT1 + j] = (unsigned)(g * HH + nbase[j] + lrow) * DD + 2 * lhalf;
    }

    v8f zero = {0.f, 0.f, 0.f, 0.f, 0.f, 0.f, 0.f, 0.f};
    v8f acc[3 * NT1 * MT1];                     // [(g*NT1+j)*MT1 + m]
#pragma unroll
    for (int q = 0; q < 3 * NT1 * MT1; ++q) acc[q] = zero;

    v2f aC[MT1], bC[3 * NT1], aN[MT1], bN[3 * NT1];
#pragma unroll
    for (int m = 0; m < MT1; ++m) aC[m] = *(const v2f*)(Ain + aoff[m]);
#pragma unroll
    for (int q = 0; q < 3 * NT1; ++q) bC[q] = *(const v2f*)(Wp + boff[q]);

#pragma unroll 4
    for (int k0 = 0; k0 < DD - 4; k0 += 4) {
        // prefetch next k-step
#pragma unroll
        for (int m = 0; m < MT1; ++m) aN[m] = *(const v2f*)(Ain + aoff[m] + k0 + 4);
#pragma unroll
        for (int q = 0; q < 3 * NT1; ++q) bN[q] = *(const v2f*)(Wp + boff[q] + k0 + 4);
        // consume current
#pragma unroll
        for (int q = 0; q < 3 * NT1; ++q)
#pragma unroll
            for (int m = 0; m < MT1; ++m)
                acc[q * MT1 + m] = __builtin_amdgcn_wmma_f32_16x16x4_f32(
                    false, aC[m], false, bC[q], (short)0, acc[q * MT1 + m], false, false);
#pragma unroll
        for (int m = 0; m < MT1; ++m) aC[m] = aN[m];
#pragma unroll
        for (int q = 0; q < 3 * NT1; ++q) bC[q] = bN[q];
    }
#pragma unroll
    for (int q = 0; q < 3 * NT1; ++q)
#pragma unroll
        for (int m = 0; m < MT1; ++m)
            acc[q * MT1 + m] = __builtin_amdgcn_wmma_f32_16x16x4_f32(
                false, aC[m], false, bC[q], (short)0, acc[q * MT1 + m], false, false);

    // Gate epilogue: diff = sp(-f)-sp(-i); log_f=-sp(diff); log_c=-sp(-diff)+log_g(h)
#pragma unroll
    for (int m = 0; m < MT1; ++m) {
#pragma unroll
        for (int j = 0; j < NT1; ++j) {
            v8f aH = acc[(0 * NT1 + j) * MT1 + m];
            v8f aI = acc[(1 * NT1 + j) * MT1 + m];
            v8f aF = acc[(2 * NT1 + j) * MT1 + m];
#pragma unroll
            for (int i = 0; i < 8; ++i) {
                float diff = softplusf(-aF[i]) - softplusf(-aI[i]);
                float lf = -softplusf(diff);
                float lc = -softplusf(-diff) + log_g(aH[i]);
                unsigned r = (unsigned)(row0 + m * 16 + i + 8 * lhalf);
                unsigned idx = r * HH + (unsigned)(nbase[j] + lrow);
                wsF[idx] = lf;
                wsC[idx] = lc;
            }
        }
    }
}

// ---------------------------------------------------------------------------
// Scan phase 1: per-chunk reduction. For each (b, chunk, c):
//   F = sum(log_f), R via R = logaddexp(log_f + R, log_c)
// ---------------------------------------------------------------------------
__global__ __launch_bounds__(256) void minlstm_scan_reduce(
    const float* __restrict__ wsF, const float* __restrict__ wsC,
    float* __restrict__ chF, float* __restrict__ chR)
{
    int t = blockIdx.x * blockDim.x + threadIdx.x;   // [0, BB*NC*HH)
    int c = t & (HH - 1);
    int rest = t >> 10;
    int chunk = rest & (NC - 1);
    int b = rest >> 6;

    size_t base = ((size_t)b * SS + (size_t)chunk * CSZ) * HH + c;
    float F = 0.0f;
    float R = -__builtin_inff();
    for (int s = 0; s < CSZ; ++s) {
        float lf = wsF[base];
        float lc = wsC[base];
        R = logaddexpf_(lf + R, lc);
        F += lf;
        base += HH;
    }
    size_t o = ((size_t)b * NC + chunk) * HH + c;
    chF[o] = F;
    chR[o] = R;
}

// ---------------------------------------------------------------------------
// Scan phase 2: sequential combine across NC chunk summaries per channel.
// ---------------------------------------------------------------------------
__global__ __launch_bounds__(256) void minlstm_scan_combine(
    const float* __restrict__ prev_hidden,
    const float* __restrict__ chF, const float* __restrict__ chR,
    float* __restrict__ chS)
{
    int t = blockIdx.x * blockDim.x + threadIdx.x;   // [0, BB*HH)
    int c = t & (HH - 1);
    int b = t >> 10;

    float run = log_g(prev_hidden[(size_t)b * HH + c]);
    for (int chunk = 0; chunk < NC; ++chunk) {
        size_t o = ((size_t)b * NC + chunk) * HH + c;
        chS[o] = run;
        run = logaddexpf_(chF[o] + run, chR[o]);
    }
}

// ---------------------------------------------------------------------------
// Scan phase 3: replay each chunk from its entry state, emit h = exp(log_h)
// in place over wsC; last chunk writes new_hidden.
// ---------------------------------------------------------------------------
__global__ __launch_bounds__(256) void minlstm_scan_emit(
    const float* __restrict__ wsF, float* __restrict__ wsC,
    const float* __restrict__ chS, float* __restrict__ new_hidden)
{
    int t = blockIdx.x * blockDim.x + threadIdx.x;   // [0, BB*NC*HH)
    int c = t & (HH - 1);
    int rest = t >> 10;
    int chunk = rest & (NC - 1);
    int b = rest >> 6;

    float lh = chS[((size_t)b * NC + chunk) * HH + c];
    size_t base = ((size_t)b * SS + (size_t)chunk * CSZ) * HH + c;
    for (int s = 0; s < CSZ; ++s) {
        float lf = wsF[base];
        float lc = wsC[base];
        lh = logaddexpf_(lf + lh, lc);
        wsC[base] = expf(lh);        // read-before-write per element
        base += HH;
    }
    if (chunk == NC - 1)
        new_hidden[(size_t)b * HH + c] = expf(lh);
}

// ---------------------------------------------------------------------------
// Kernel 2: outputs = h @ W_out^T.  Per wave: MT2=2 row tiles x NT2=4 col
// tiles -> 8 accumulators, 8 WMMAs per k-step fed by 2 A + 4 B loads.
// ---------------------------------------------------------------------------
#define MT2 2
#define NT2 4
__global__ __launch_bounds__(128) void minlstm_gemm2_out(
    const float* __restrict__ Hmat,  // (MM, HH)
    const float* __restrict__ Wo,    // (DD, HH)
    float* __restrict__ Out)         // (MM, DD)
{
    const int lane  = threadIdx.x & 31;
    const int wave  = blockIdx.x * (blockDim.x >> 5) + (threadIdx.x >> 5);
    const int rowGrp = wave & (MM / (16 * MT2) - 1);   // 1024 row groups
    const int dg     = wave >> 10;                     // 0..7
    const int row0  = rowGrp * (16 * MT2);
    const int lrow  = lane & 15;
    const int lhalf = lane >> 4;

    unsigned aoff[MT2];
#pragma unroll
    for (int m = 0; m < MT2; ++m)
        aoff[m] = (unsigned)(row0 + m * 16 + lrow) * HH + 2 * lhalf;

    int dbase[NT2];
    unsigned boff[NT2];
#pragma unroll
    for (int j = 0; j < NT2; ++j) {
        dbase[j] = dg * (16 * NT2) + j * 16;
        boff[j] = (unsigned)(dbase[j] + lrow) * HH + 2 * lhalf;
    }

    v8f zero = {0.f, 0.f, 0.f, 0.f, 0.f, 0.f, 0.f, 0.f};
    v8f acc[NT2 * MT2];                    // [j*MT2 + m]
#pragma unroll
    for (int q = 0; q < NT2 * MT2; ++q) acc[q] = zero;

    v2f aC[MT2], bC[NT2], aN[MT2], bN[NT2];
#pragma unroll
    for (int m = 0; m < MT2; ++m) aC[m] = *(const v2f*)(Hmat + aoff[m]);
#pragma unroll
    for (int j = 0; j < NT2; ++j) bC[j] = *(const v2f*)(Wo + boff[j]);

#pragma unroll 4
    for (int k0 = 0; k0 < HH - 4; k0 += 4) {
#pragma unroll
        for (int m = 0; m < MT2; ++m) aN[m] = *(const v2f*)(Hmat + aoff[m] + k0 + 4);
#pragma unroll
        for (int j = 0; j < NT2; ++j) bN[j] = *(const v2f*)(Wo + boff[j] + k0 + 4);
#pragma unroll
        for (int j = 0; j < NT2; ++j)
#pragma unroll
            for (int m = 0; m < MT2; ++m)
                acc[j * MT2 + m] = __builtin_amdgcn_wmma_f32_16x16x4_f32(
                    false, aC[m], false, bC[j], (short)0, acc[j * MT2 + m], false, false);
#pragma unroll
        for (int m = 0; m < MT2; ++m) aC[m] = aN[m];
#pragma unroll
        for (int j = 0; j < NT2; ++j) bC[j] = bN[j];
    }
#pragma unroll
    for (int j = 0; j < NT2; ++j)
#pragma unroll
        for (int m = 0; m < MT2; ++m)
            acc[j * MT2 + m] = __builtin_amdgcn_wmma_f32_16x16x4_f32(
                false, aC[m], false, bC[j], (short)0, acc[j * MT2 + m], false, false);

#pragma unroll
    for (int m = 0; m < MT2; ++m)
#pragma unroll
        for (int j = 0; j < NT2; ++j)
#pragma unroll
            for (int i = 0; i < 8; ++i) {
                unsigned r = (unsigned)(row0 + m * 16 + i + 8 * lhalf);
                Out[r * DD + (unsigned)(dbase[j] + lrow)] = acc[j * MT2 + m][i];
            }
}

// ---------------------------------------------------------------------------
extern "C" void kernel_launch(void* const* d_in, const int* in_sizes, int n_in,
                              void* d_out, int out_size, void* d_ws, size_t ws_size,
                              hipStream_t stream) {
    const float* inputs      = (const float*)d_in[0];  // (B,S,D)
    const float* prev_hidden = (const float*)d_in[1];  // (B,1,H)
    const float* W_proj      = (const float*)d_in[2];  // (3H,D)
    const float* W_out       = (const float*)d_in[3];  // (D,H)

    float* out    = (float*)d_out;                 // (B,S,D) flat
    float* nh_out = out + (size_t)MM * DD;         // (B,1,H) flat tail

    // Workspace layout (floats)
    float* wsF = (float*)d_ws;                     // MM*HH  log_forget
    float* wsC = wsF + (size_t)MM * HH;            // MM*HH  log_candidate -> h
    float* chF = wsC + (size_t)MM * HH;            // BB*NC*HH
    float* chR = chF + (size_t)BB * NC * HH;       // BB*NC*HH
    float* chS = chR + (size_t)BB * NC * HH;       // BB*NC*HH

    // GEMM1 + gates: 1024 row groups * 32 col groups = 32768 waves, 4/block
    minlstm_gemm1_gates<<<(MM / (16 * MT1)) * (HH / (16 * NT1)) / 4, 128, 0, stream>>>(
        inputs, W_proj, wsF, wsC);

    // Chunked parallel scan
    const int nscan = BB * NC * HH;                // 524288 threads
    minlstm_scan_reduce<<<nscan / 256, 256, 0, stream>>>(wsF, wsC, chF, chR);
    minlstm_scan_combine<<<(BB * HH) / 256, 256, 0, stream>>>(prev_hidden, chF, chR, chS);
    minlstm_scan_emit<<<nscan / 256, 256, 0, stream>>>(wsF, wsC, chS, nh_out);

    // GEMM2: 1024 row groups * 8 col groups = 8192 waves, 4/block
    minlstm_gemm2_out<<<(MM / (16 * MT2)) * (DD / (16 * NT2)) / 4, 128, 0, stream>>>(
        wsC, W_out, out);
}